// SVTRStage_81827716923509
// MI455X (gfx1250) — compile-verified
//
#include <hip/hip_runtime.h>
#include <math.h>

typedef __attribute__((ext_vector_type(2))) float v2f;
typedef __attribute__((ext_vector_type(8))) float v8f;

#define DIM_    128
#define HEADS_  4
#define HD_     32
#define NSEQ    2048
#define BATCH   4
#define HH_     16
#define WW_     128
#define HIDDEN_ 512
#define OUTD_   256
#define ROWS    (BATCH * NSEQ)    // 8192
#define CROWS   (BATCH * 8 * WW_) // 4096

#define WMMA_F32(a, b, c) \
    __builtin_amdgcn_wmma_f32_16x16x4_f32(false, (a), false, (b), (short)0, (c), false, false)

// ---------------------------------------------------------------------------
// f32 WMMA GEMM: C[M,N] = A[M,K] @ B[K,N] (+bias) (+resid) (opt exact GELU).
// One wave -> 16x64 tile (4 accumulators): each A fragment feeds 4 WMMAs.
// ---------------------------------------------------------------------------
__global__ void gemm_wmma_f32(const float* __restrict__ A,
                              const float* __restrict__ Bw,
                              const float* __restrict__ bias,
                              const float* __restrict__ resid,
                              float* __restrict__ C,
                              int M, int N, int K, int act)
{
    int wave = (int)((blockIdx.x * blockDim.x + threadIdx.x) >> 5);
    int lane = threadIdx.x & 31;
    int ntn  = N >> 6;                      // 64-wide N tiles
    int total = (M >> 4) * ntn;
    if (wave >= total) return;              // wave-uniform: EXEC stays all-1s
    int mt = wave / ntn, nt = wave % ntn;
    int half = lane >> 4;
    int l16  = lane & 15;
    int arow  = (mt << 4) + l16;            // A-fragment row owned by this lane
    int ncol0 = (nt << 6) + l16;            // first of 4 owned B/C columns
    const float* __restrict__ arp = A + (size_t)arow * K;

    v8f c0 = {0.f,0.f,0.f,0.f,0.f,0.f,0.f,0.f};
    v8f c1 = c0, c2 = c0, c3 = c0;
    for (int k0 = 0; k0 < K; k0 += 4) {
        int ka = k0 + (half << 1);
        v2f a;
        a.x = arp[ka];
        a.y = arp[ka + 1];
        const float* __restrict__ bp = Bw + (size_t)ka * N + ncol0;
        v2f b0, b1, b2, b3;
        b0.x = bp[0];  b0.y = bp[N];
        b1.x = bp[16]; b1.y = bp[N + 16];
        b2.x = bp[32]; b2.y = bp[N + 32];
        b3.x = bp[48]; b3.y = bp[N + 48];
        c0 = WMMA_F32(a, b0, c0);
        c1 = WMMA_F32(a, b1, c1);
        c2 = WMMA_F32(a, b2, c2);
        c3 = WMMA_F32(a, b3, c3);
    }

    float bv[4] = {0.f, 0.f, 0.f, 0.f};
    if (bias) {
#pragma unroll
        for (int j = 0; j < 4; ++j) bv[j] = bias[ncol0 + 16 * j];
    }
#pragma unroll
    for (int r = 0; r < 8; ++r) {
        int m = (mt << 4) + r + (half << 3);
        float vj[4] = {c0[r], c1[r], c2[r], c3[r]};
#pragma unroll
        for (int j = 0; j < 4; ++j) {
            int col = ncol0 + 16 * j;
            float v = vj[j] + bv[j];
            if (resid) v += resid[(size_t)m * N + col];
            if (act == 1) v = 0.5f * v * (1.f + erff(v * 0.70710678118654752f));
            C[(size_t)m * N + col] = v;
        }
    }
}

// ---------------------------------------------------------------------------
// Local masked attention: window |dh|<=3, |dw|<=5 (<=77 keys per query).
// One thread per (batch, query, head); online (flash-style) softmax.
// qkv layout per token: [q(128) | k(128) | v(128)], head-major inside each.
// ---------------------------------------------------------------------------
__global__ void attn_local(const float* __restrict__ qkv, float* __restrict__ o)
{
    int idx = blockIdx.x * blockDim.x + threadIdx.x;
    if (idx >= BATCH * NSEQ * HEADS_) return;
    int head = idx & 3;
    int n    = (idx >> 2) & (NSEQ - 1);
    int b    = idx >> 13;
    int hq = n >> 7, wq = n & 127;

    const float* __restrict__ qp = qkv + (size_t)(b * NSEQ + n) * (3 * DIM_) + head * HD_;
    float q[HD_];
#pragma unroll
    for (int d = 0; d < HD_; ++d) q[d] = qp[d];

    const float scale = 0.17677669529663687f;   // 1/sqrt(32)
    float mrun = -1e30f, lrun = 0.f;
    float acc[HD_];
#pragma unroll
    for (int d = 0; d < HD_; ++d) acc[d] = 0.f;

    for (int dh = -3; dh <= 3; ++dh) {
        int hk = hq + dh;
        if (hk < 0 || hk >= HH_) continue;
        for (int dw = -5; dw <= 5; ++dw) {
            int wk = wq + dw;
            if (wk < 0 || wk >= WW_) continue;
            int mk = (hk << 7) + wk;
            const float* __restrict__ kp =
                qkv + (size_t)(b * NSEQ + mk) * (3 * DIM_) + DIM_ + head * HD_;
            const float* __restrict__ vp = kp + DIM_;
            float s = 0.f;
#pragma unroll
            for (int d = 0; d < HD_; ++d) s += q[d] * kp[d];
            s *= scale;
            float nm = fmaxf(mrun, s);
            float f  = __expf(mrun - nm);
            float p  = __expf(s - nm);
            lrun = lrun * f + p;
#pragma unroll
            for (int d = 0; d < HD_; ++d) acc[d] = acc[d] * f + p * vp[d];
            mrun = nm;
        }
    }
    float inv = 1.f / lrun;
    float* __restrict__ op = o + (size_t)(b * NSEQ + n) * DIM_ + head * HD_;
#pragma unroll
    for (int d = 0; d < HD_; ++d) op[d] = acc[d] * inv;
}

// ---------------------------------------------------------------------------
// LayerNorm: one wave32 per row, D/32 elements per lane, shfl_xor reduction.
// ---------------------------------------------------------------------------
template <int D>
__global__ void ln_rows(const float* __restrict__ X, const float* __restrict__ g,
                        const float* __restrict__ bta, float* __restrict__ Y, int rows)
{
    int wave = (int)((blockIdx.x * blockDim.x + threadIdx.x) >> 5);
    int lane = threadIdx.x & 31;
    if (wave >= rows) return;
    constexpr int PT = D / 32;
    const float* __restrict__ xp = X + (size_t)wave * D;
    float v[PT];
    float s = 0.f, s2 = 0.f;
#pragma unroll
    for (int i = 0; i < PT; ++i) {
        v[i] = xp[lane + 32 * i];
        s += v[i];
        s2 += v[i] * v[i];
    }
#pragma unroll
    for (int off = 16; off > 0; off >>= 1) {
        s  += __shfl_xor(s, off, 32);
        s2 += __shfl_xor(s2, off, 32);
    }
    float mean = s / (float)D;
    float var  = s2 / (float)D - mean * mean;
    float r    = rsqrtf(var + 1e-6f);
    float* __restrict__ yp = Y + (size_t)wave * D;
#pragma unroll
    for (int i = 0; i < PT; ++i) {
        int cidx = lane + 32 * i;
        yp[cidx] = (v[i] - mean) * r * g[cidx] + bta[cidx];
    }
}

// ---------------------------------------------------------------------------
// 3x3 stride-(2,1) pad-1 conv as implicit-im2col WMMA GEMM:
// M = 4096 (b,h2,w2), N = 256 out channels, K = 9*128 with gathered A.
// One wave -> 16x64 tile; k ordered (kh*3+kw)*128+ic so lane pairs share (kh,kw).
// ---------------------------------------------------------------------------
__global__ void conv_ds_wmma(const float* __restrict__ X,
                             const float* __restrict__ Wt,   // (oc, ic, kh, kw)
                             const float* __restrict__ bias,
                             float* __restrict__ Y)
{
    const int M = CROWS, N = OUTD_;
    const int KW9 = 9 * DIM_;               // weight K stride per oc = 1152
    int wave = (int)((blockIdx.x * blockDim.x + threadIdx.x) >> 5);
    int lane = threadIdx.x & 31;
    int ntn  = N >> 6;
    int total = (M >> 4) * ntn;
    if (wave >= total) return;
    int mt = wave / ntn, nt = wave % ntn;
    int half = lane >> 4, l16 = lane & 15;
    int arow = (mt << 4) + l16;
    int b  = arow >> 10;
    int h2 = (arow >> 7) & 7;
    int w2 = arow & 127;
    int ncol0 = (nt << 6) + l16;

    v8f c0 = {0.f,0.f,0.f,0.f,0.f,0.f,0.f,0.f};
    v8f c1 = c0, c2 = c0, c3 = c0;
    for (int k0 = 0; k0 < KW9; k0 += 4) {
        int k    = k0 + (half << 1);
        int khkw = k >> 7;            // 0..8
        int ic   = k & 127;           // even
        int kh = khkw / 3, kw = khkw % 3;
        int hin = 2 * h2 + kh - 1;
        int win = w2 + kw - 1;
        v2f a;
        bool inb = (hin >= 0) & (hin < HH_) & (win >= 0) & (win < WW_);
        if (inb) {
            const float* __restrict__ xp =
                X + ((size_t)b * NSEQ + hin * WW_ + win) * DIM_ + ic;
            a.x = xp[0];
            a.y = xp[1];
        } else {
            a.x = 0.f;
            a.y = 0.f;
        }
        const float* __restrict__ wp =
            Wt + ((size_t)ncol0 * DIM_ + ic) * 9 + kh * 3 + kw;
        v2f b0, b1, b2, b3;
        b0.x = wp[0];             b0.y = wp[9];
        b1.x = wp[16 * KW9];      b1.y = wp[16 * KW9 + 9];
        b2.x = wp[32 * KW9];      b2.y = wp[32 * KW9 + 9];
        b3.x = wp[48 * KW9];      b3.y = wp[48 * KW9 + 9];
        c0 = WMMA_F32(a, b0, c0);
        c1 = WMMA_F32(a, b1, c1);
        c2 = WMMA_F32(a, b2, c2);
        c3 = WMMA_F32(a, b3, c3);
    }
    float bv[4];
#pragma unroll
    for (int j = 0; j < 4; ++j) bv[j] = bias[ncol0 + 16 * j];
#pragma unroll
    for (int r = 0; r < 8; ++r) {
        int m = (mt << 4) + r + (half << 3);
        float vj[4] = {c0[r], c1[r], c2[r], c3[r]};
#pragma unroll
        for (int j = 0; j < 4; ++j)
            Y[(size_t)m * N + ncol0 + 16 * j] = vj[j] + bv[j];
    }
}

__global__ void write_tail(int* __restrict__ p)
{
    p[0] = 8;      // H2
    p[1] = WW_;    // W2
}

// ---------------------------------------------------------------------------
extern "C" void kernel_launch(void* const* d_in, const int* in_sizes, int n_in,
                              void* d_out, int out_size, void* d_ws, size_t ws_size,
                              hipStream_t stream)
{
    const float* x_in   = (const float*)d_in[0];
    // d_in[1] = mask: recomputed analytically (local 7x11 window), unused
    const float* qkv_w  = (const float*)d_in[2];
    const float* qkv_b  = (const float*)d_in[3];
    const float* proj_w = (const float*)d_in[4];
    const float* proj_b = (const float*)d_in[5];
    const float* ln1_g  = (const float*)d_in[6];
    const float* ln1_b  = (const float*)d_in[7];
    const float* ln2_g  = (const float*)d_in[8];
    const float* ln2_b  = (const float*)d_in[9];
    const float* fc1_w  = (const float*)d_in[10];
    const float* fc1_b  = (const float*)d_in[11];
    const float* fc2_w  = (const float*)d_in[12];
    const float* fc2_b  = (const float*)d_in[13];
    const float* ds_w   = (const float*)d_in[14];
    const float* ds_b   = (const float*)d_in[15];
    const float* sn_g   = (const float*)d_in[16];
    const float* sn_b   = (const float*)d_in[17];

    // Workspace layout (floats), aliased; total footprint 24 MB.
    float* ws   = (float*)d_ws;
    float* xbuf = ws;                              // 8192*128
    float* qkvb = xbuf + (size_t)ROWS * DIM_;      // 8192*384
    float* obuf = qkvb + (size_t)ROWS * 3 * DIM_;  // 8192*128
    float* tbuf = obuf + (size_t)ROWS * DIM_;      // 8192*128
    float* hbuf = qkvb;   // 8192*512 overlays qkvb+obuf exactly
    float* cbuf = qkvb;   // 4096*256 conv output (after last layer)

    hipMemcpyAsync(xbuf, x_in, (size_t)ROWS * DIM_ * sizeof(float),
                   hipMemcpyDeviceToDevice, stream);

    auto gemmL = [&](const float* A, const float* Bw, const float* bias,
                     const float* resid, float* C, int M, int N, int K, int act) {
        int tiles  = (M / 16) * (N / 64);
        int blocks = (tiles + 7) / 8;       // 8 waves (16x64 tiles) per block
        gemm_wmma_f32<<<blocks, 256, 0, stream>>>(A, Bw, bias, resid, C, M, N, K, act);
    };

    for (int l = 0; l < 3; ++l) {
        gemmL(xbuf, qkv_w + (size_t)l * DIM_ * 3 * DIM_, qkv_b + l * 3 * DIM_,
              nullptr, qkvb, ROWS, 3 * DIM_, DIM_, 0);

        attn_local<<<(BATCH * NSEQ * HEADS_) / 256, 256, 0, stream>>>(qkvb, obuf);

        gemmL(obuf, proj_w + (size_t)l * DIM_ * DIM_, proj_b + l * DIM_,
              xbuf, tbuf, ROWS, DIM_, DIM_, 0);
        ln_rows<128><<<ROWS / 8, 256, 0, stream>>>(tbuf, ln1_g + l * DIM_,
                                                   ln1_b + l * DIM_, xbuf, ROWS);

        gemmL(xbuf, fc1_w + (size_t)l * DIM_ * HIDDEN_, fc1_b + l * HIDDEN_,
              nullptr, hbuf, ROWS, HIDDEN_, DIM_, 1);
        gemmL(hbuf, fc2_w + (size_t)l * HIDDEN_ * DIM_, fc2_b + l * DIM_,
              xbuf, tbuf, ROWS, DIM_, HIDDEN_, 0);
        ln_rows<128><<<ROWS / 8, 256, 0, stream>>>(tbuf, ln2_g + l * DIM_,
                                                   ln2_b + l * DIM_, xbuf, ROWS);
    }

    {
        int tiles = (CROWS / 16) * (OUTD_ / 64);
        conv_ds_wmma<<<(tiles + 7) / 8, 256, 0, stream>>>(xbuf, ds_w, ds_b, cbuf);
    }
    ln_rows<256><<<CROWS / 8, 256, 0, stream>>>(cbuf, sn_g, sn_b, (float*)d_out, CROWS);
    write_tail<<<1, 1, 0, stream>>>((int*)d_out + (size_t)CROWS * OUTD_);
}